// GraphSPADE_88510686036718
// MI455X (gfx1250) — compile-verified
//
#include <hip/hip_runtime.h>
#include <hip/hip_bf16.h>

#define N_NODES 100000
#define N_EDGES 1600000
#define IN_CH   128
#define MASK_CH 16
#define HID     128
#define BN_EPS  1e-5f

typedef float v2f __attribute__((ext_vector_type(2)));
typedef float v8f __attribute__((ext_vector_type(8)));

// ---------------------------------------------------------------------------
// Workspace layout (floats):
//   deg      [N]          offset 0
//   dinv     [N]          offset N
//   agg_mask [N*16]       offset 2N
//   mask_feat[N*128]      offset 18N
//   agg_mf   [N*128]      offset 146N
//   stats    [512]        offset 274N   (sum[128], sumsq[128], mu[128], istd[128])
// ---------------------------------------------------------------------------

// deg = 1.0 (self loop), zero BN accumulators
__global__ void k_init(float* __restrict__ deg, float* __restrict__ stats) {
    int i = blockIdx.x * blockDim.x + threadIdx.x;
    if (i < N_NODES) deg[i] = 1.0f;
    if (i < 256) stats[i] = 0.0f;
}

// deg[dst] += 1 per edge
__global__ void k_degree(const int* __restrict__ dst, float* __restrict__ deg) {
    int e = blockIdx.x * blockDim.x + threadIdx.x;
    if (e < N_EDGES) atomicAdd(&deg[dst[e]], 1.0f);
}

// dinv = rsqrt(deg); agg_mask = dinv^2 * mask  (self-loop term, plain write)
__global__ void k_dinv_selfloop_mask(const float* __restrict__ deg,
                                     const float* __restrict__ mask,
                                     float* __restrict__ dinv,
                                     float* __restrict__ agg_mask) {
    int t = blockIdx.x * blockDim.x + threadIdx.x;   // over N*16
    if (t >= N_NODES * MASK_CH) return;
    int node = t >> 4;
    float w = __frsqrt_rn(deg[node]);
    if ((t & 15) == 0) dinv[node] = w;
    agg_mask[t] = w * w * mask[t];
}

// agg_mask[d] += dinv[s]*dinv[d] * mask[s]   (one thread per edge-channel)
__global__ void k_edge_agg_mask(const int* __restrict__ src, const int* __restrict__ dst,
                                const float* __restrict__ dinv,
                                const float* __restrict__ mask,
                                float* __restrict__ agg_mask) {
    long long t = (long long)blockIdx.x * blockDim.x + threadIdx.x;  // E*16
    if (t >= (long long)N_EDGES * MASK_CH) return;
    int e = (int)(t >> 4);
    int c = (int)(t & 15);
    int s = src[e], d = dst[e];
    float w = dinv[s] * dinv[d];
    atomicAdd(&agg_mask[d * MASK_CH + c], mask[s * MASK_CH + c] * w);
}

// mask_feat = relu(agg_mask[N,16] @ W1^T[16,128] + b1)   via V_WMMA_F32_16X16X4_F32
// Block = 256 thr = 8 waves; wave w owns column tile w; blockIdx.x = 16-row stripe.
__global__ void k_gemm1_relu(const float* __restrict__ agg_mask,
                             const float* __restrict__ W1,   // [128,16] row-major
                             const float* __restrict__ b1,
                             float* __restrict__ mask_feat) {
    const int lane = threadIdx.x & 31;
    const int wave = threadIdx.x >> 5;
    const int rb = blockIdx.x * 16;        // row base (node)
    const int cb = wave * 16;              // col base (hidden channel)
    const int n  = lane & 15;              // also m for A loads
    const int kh = (lane >> 4) * 2;        // K sub-offset for this half-wave
    const int mbase = (lane >> 4) * 8;     // accumulator row base

    v8f acc = {};
    #pragma unroll
    for (int k0 = 0; k0 < MASK_CH; k0 += 4) {
        v2f a, b;
        a.x = agg_mask[(rb + n) * MASK_CH + k0 + kh];
        a.y = agg_mask[(rb + n) * MASK_CH + k0 + kh + 1];
        b.x = W1[(cb + n) * MASK_CH + k0 + kh];      // B[k][n] = W1[n][k]
        b.y = W1[(cb + n) * MASK_CH + k0 + kh + 1];
        acc = __builtin_amdgcn_wmma_f32_16x16x4_f32(false, a, false, b,
                                                    (short)0, acc, false, false);
    }
    float bias = b1[cb + n];
    #pragma unroll
    for (int v = 0; v < 8; ++v) {
        int row = rb + mbase + v;
        mask_feat[row * HID + cb + n] = fmaxf(acc[v] + bias, 0.0f);
    }
}

// agg_mf = dinv^2 * mask_feat  (self-loop term, plain write)
__global__ void k_selfloop_mf(const float* __restrict__ dinv,
                              const float* __restrict__ mask_feat,
                              float* __restrict__ agg_mf) {
    long long t = (long long)blockIdx.x * blockDim.x + threadIdx.x;  // N*128
    if (t >= (long long)N_NODES * HID) return;
    int node = (int)(t >> 7);
    float w = dinv[node];
    agg_mf[t] = w * w * mask_feat[t];
}

// agg_mf[d] += dinv[s]*dinv[d] * mask_feat[s]  : one wave per edge, float4 per lane
__global__ void k_edge_agg_mf(const int* __restrict__ src, const int* __restrict__ dst,
                              const float* __restrict__ dinv,
                              const float* __restrict__ mask_feat,
                              float* __restrict__ agg_mf) {
    int lane = threadIdx.x & 31;
    int e = (blockIdx.x * blockDim.x + threadIdx.x) >> 5;
    if (e >= N_EDGES) return;
    int s = src[e], d = dst[e];
    float w = dinv[s] * dinv[d];
    const float4* mf4 = (const float4*)(mask_feat + (long long)s * HID);
    float4 v = mf4[lane];
    float* outp = agg_mf + (long long)d * HID + lane * 4;
    atomicAdd(outp + 0, v.x * w);
    atomicAdd(outp + 1, v.y * w);
    atomicAdd(outp + 2, v.z * w);
    atomicAdd(outp + 3, v.w * w);
}

// Batch-norm stats: per-channel sum and sum of squares (grid-stride, stride%128==0)
__global__ void k_bn_stats(const float* __restrict__ x,
                           float* __restrict__ gsum, float* __restrict__ gsumsq) {
    __shared__ float ss[256], sq[256];
    long long total = (long long)N_NODES * IN_CH;
    long long start = (long long)blockIdx.x * blockDim.x + threadIdx.x;
    long long stride = (long long)gridDim.x * blockDim.x;   // multiple of 128
    float s = 0.0f, q = 0.0f;
    for (long long i = start; i < total; i += stride) {
        float v = x[i];
        s += v; q += v * v;
    }
    ss[threadIdx.x] = s; sq[threadIdx.x] = q;
    __syncthreads();
    if (threadIdx.x < 128) {
        // threadIdx.x and threadIdx.x+128 cover the same channel (stride%128==0)
        float ts = ss[threadIdx.x] + ss[threadIdx.x + 128];
        float tq = sq[threadIdx.x] + sq[threadIdx.x + 128];
        atomicAdd(&gsum[threadIdx.x], ts);
        atomicAdd(&gsumsq[threadIdx.x], tq);
    }
}

__global__ void k_bn_final(const float* __restrict__ gsum, const float* __restrict__ gsumsq,
                           float* __restrict__ mu, float* __restrict__ istd) {
    int c = threadIdx.x;
    if (c < IN_CH) {
        float m = gsum[c] * (1.0f / N_NODES);
        float var = gsumsq[c] * (1.0f / N_NODES) - m * m;
        mu[c] = m;
        istd[c] = __frsqrt_rn(var + BN_EPS);
    }
}

// gamma/beta GEMMs fused with output: out = (x-mu)*istd*(1+gamma) + beta
// gamma = agg_mf @ Wg^T + bg ; beta = agg_mf @ Wb^T + bb ; K=128 in steps of 4.
__global__ void k_gemm2_fused_out(const float* __restrict__ agg_mf,
                                  const float* __restrict__ Wg, const float* __restrict__ bg,
                                  const float* __restrict__ Wb, const float* __restrict__ bb,
                                  const float* __restrict__ x,
                                  const float* __restrict__ mu, const float* __restrict__ istd,
                                  float* __restrict__ out) {
    const int lane = threadIdx.x & 31;
    const int wave = threadIdx.x >> 5;
    const int rb = blockIdx.x * 16;
    const int cb = wave * 16;
    const int n  = lane & 15;
    const int kh = (lane >> 4) * 2;
    const int mbase = (lane >> 4) * 8;

    v8f accg = {};
    v8f accb = {};
    const float* arow = agg_mf + (long long)(rb + n) * HID;
    const float* wgrow = Wg + (long long)(cb + n) * HID;
    const float* wbrow = Wb + (long long)(cb + n) * HID;
    #pragma unroll 8
    for (int k0 = 0; k0 < HID; k0 += 4) {
        v2f a, bgv, bbv;
        a.x = arow[k0 + kh];
        a.y = arow[k0 + kh + 1];
        bgv.x = wgrow[k0 + kh];
        bgv.y = wgrow[k0 + kh + 1];
        bbv.x = wbrow[k0 + kh];
        bbv.y = wbrow[k0 + kh + 1];
        accg = __builtin_amdgcn_wmma_f32_16x16x4_f32(false, a, false, bgv,
                                                     (short)0, accg, false, false);
        accb = __builtin_amdgcn_wmma_f32_16x16x4_f32(false, a, false, bbv,
                                                     (short)0, accb, false, false);
    }
    int col = cb + n;
    float bgs = bg[col], bbs = bb[col];
    float muv = mu[col], isv = istd[col];
    #pragma unroll
    for (int v = 0; v < 8; ++v) {
        int row = rb + mbase + v;
        long long idx = (long long)row * IN_CH + col;
        float xn = (x[idx] - muv) * isv;
        out[idx] = xn * (1.0f + accg[v] + bgs) + (accb[v] + bbs);
    }
}

extern "C" void kernel_launch(void* const* d_in, const int* in_sizes, int n_in,
                              void* d_out, int out_size, void* d_ws, size_t ws_size,
                              hipStream_t stream) {
    const float* x    = (const float*)d_in[0];   // [N,128]
    const float* mask = (const float*)d_in[1];   // [N,16]
    const int*   eidx = (const int*)d_in[2];     // [2,E] (int32 — JAX default int)
    const float* W1   = (const float*)d_in[3];   // [128,16]
    const float* b1   = (const float*)d_in[4];   // [128]
    const float* Wg   = (const float*)d_in[5];   // [128,128]
    const float* bg   = (const float*)d_in[6];   // [128]
    const float* Wb   = (const float*)d_in[7];   // [128,128]
    const float* bb   = (const float*)d_in[8];   // [128]
    float* out = (float*)d_out;

    const int* src = eidx;
    const int* dst = eidx + N_EDGES;

    float* ws = (float*)d_ws;
    float* deg       = ws;
    float* dinv      = ws + (long long)N_NODES;
    float* agg_mask  = ws + 2LL * N_NODES;
    float* mask_feat = ws + 18LL * N_NODES;
    float* agg_mf    = ws + 146LL * N_NODES;
    float* stats     = ws + 274LL * N_NODES;
    float* gsum   = stats;
    float* gsumsq = stats + 128;
    float* muv    = stats + 256;
    float* istd   = stats + 384;

    // 1. init: deg=1 (self loop), BN accumulators = 0
    k_init<<<(N_NODES + 255) / 256, 256, 0, stream>>>(deg, stats);
    // 2. degrees from edges
    k_degree<<<(N_EDGES + 255) / 256, 256, 0, stream>>>(dst, deg);
    // 3. dinv + self-loop init of agg_mask
    k_dinv_selfloop_mask<<<(N_NODES * MASK_CH + 255) / 256, 256, 0, stream>>>(
        deg, mask, dinv, agg_mask);
    // 4. edge aggregation of 16-ch mask
    {
        long long tot = (long long)N_EDGES * MASK_CH;
        k_edge_agg_mask<<<(int)((tot + 255) / 256), 256, 0, stream>>>(
            src, dst, dinv, mask, agg_mask);
    }
    // 5. GEMM1 + ReLU (WMMA f32) -> mask_feat   (100000/16 = 6250 row tiles)
    k_gemm1_relu<<<N_NODES / 16, 256, 0, stream>>>(agg_mask, W1, b1, mask_feat);
    // 6. self-loop init of agg_mf
    {
        long long tot = (long long)N_NODES * HID;
        k_selfloop_mf<<<(int)((tot + 255) / 256), 256, 0, stream>>>(
            dinv, mask_feat, agg_mf);
    }
    // 7. edge aggregation of 128-ch mask_feat (1 wave / edge, 8 edges / block)
    k_edge_agg_mf<<<N_EDGES / 8, 256, 0, stream>>>(src, dst, dinv, mask_feat, agg_mf);
    // 8. batch-norm stats + finalize
    k_bn_stats<<<512, 256, 0, stream>>>(x, gsum, gsumsq);
    k_bn_final<<<1, 128, 0, stream>>>(gsum, gsumsq, muv, istd);
    // 9. gamma/beta GEMMs (WMMA f32 ×2) fused with normalization + output
    k_gemm2_fused_out<<<N_NODES / 16, 256, 0, stream>>>(
        agg_mf, Wg, bg, Wb, bb, x, muv, istd, out);
}